// Embeddings_39032662786175
// MI455X (gfx1250) — compile-verified
//
#include <hip/hip_runtime.h>
#include <hip/hip_bf16.h>

typedef __attribute__((ext_vector_type(2))) float v2f;
typedef __attribute__((ext_vector_type(8))) float v8f;

#define DIMN   768
#define KPATCH 588            // 3*14*14
#define KCH    196            // one channel's K slice (14*14)
#define LTOK   256
#define BATCH  64
#define MTOT   (BATCH * LTOK) // 16384

// ---------------------------------------------------------------------------
// Generic transpose: out[c*R + r] = in[r*C + c]   (in is R x C row-major)
// ---------------------------------------------------------------------------
__global__ void transpose_k(const float* __restrict__ in, float* __restrict__ out,
                            int R, int C) {
    int i = blockIdx.x * 256 + threadIdx.x;
    if (i >= R * C) return;
    int r = i / C;
    int c = i - r * C;
    out[c * R + r] = in[r * C + c];
}

// ---------------------------------------------------------------------------
// Patch-embed GEMM: tokens[16384 x 768] = Xpatch[16384 x 588] @ Wt[588 x 768] + b
// Block = 32 M-rows; K staged per channel (196) in LDS.
// Each wave owns a 32x96 output: 2 A frags x 6 N-slices = 12 accumulator sets,
// so each B fragment pair feeds two WMMAs (12 wmma : 12 vmem : 2 ds per k-step).
// ---------------------------------------------------------------------------
__global__ __launch_bounds__(256)
void patch_gemm(const float* __restrict__ x, const float* __restrict__ Wt,
                const float* __restrict__ bias, float* __restrict__ tokens) {
    __shared__ float As[32][KCH + 8];      // stride 204 floats: conflict-free ds_load_b64

    const int m0 = blockIdx.x * 32;

    const int lane  = threadIdx.x & 31;
    const int wave  = threadIdx.x >> 5;
    const int mrow  = lane & 15;           // A: row within the 16x4 fragment
    const int khalf = (lane >> 4) << 1;    // A/B: K offset 0 (lanes 0-15) or 2 (16-31)
    const int ncol  = lane & 15;           // B/C/D: column within 16-wide slice
    const int mhalf = (lane >> 4) << 3;    // C/D: +8 M rows for lanes 16-31

    v8f acc0[6] = {};                      // rows m0 .. m0+15
    v8f acc1[6] = {};                      // rows m0+16 .. m0+31

    for (int ch = 0; ch < 3; ch++) {       // chunk == input channel
        __syncthreads();                   // previous chunk fully consumed
        for (int i = threadIdx.x; i < 32 * KCH; i += 256) {
            int row = i / KCH;             // 0..31
            int kc  = i - row * KCH;       // 0..195  (= p*14 + q)
            int m   = m0 + row;
            int b   = m >> 8;
            int rem = m & 255;
            int h   = rem >> 4;
            int w   = rem & 15;
            int p   = kc / 14;
            int q   = kc - p * 14;
            As[row][kc] = x[((size_t)(b * 3 + ch) * 224 + h * 14 + p) * 224 + (w * 14 + q)];
        }
        __syncthreads();

        const int kbase = ch * KCH;
        for (int k0 = 0; k0 < KCH; k0 += 4) {
            v2f a0 = *(const v2f*)&As[mrow][k0 + khalf];        // ds_load_b64
            v2f a1 = *(const v2f*)&As[mrow + 16][k0 + khalf];
            const float* brow = Wt + (size_t)(kbase + k0 + khalf) * DIMN + wave * 16 + ncol;
#pragma unroll
            for (int j = 0; j < 6; j++) {
                v2f b;
                b.x = brow[j * 128];
                b.y = brow[j * 128 + DIMN];
                acc0[j] = __builtin_amdgcn_wmma_f32_16x16x4_f32(
                    false, a0, false, b, (short)0, acc0[j], false, false);
                acc1[j] = __builtin_amdgcn_wmma_f32_16x16x4_f32(
                    false, a1, false, b, (short)0, acc1[j], false, false);
            }
        }
    }

#pragma unroll
    for (int j = 0; j < 6; j++) {
        int n    = j * 128 + wave * 16 + ncol;
        float bv = bias ? bias[n] : 0.0f;
#pragma unroll
        for (int r = 0; r < 8; r++) {
            int m = m0 + r + mhalf;
            tokens[(size_t)m * DIMN + n]        = acc0[j][r] + bv;
            tokens[(size_t)(m + 16) * DIMN + n] = acc1[j][r] + bv;
        }
    }
}

// ---------------------------------------------------------------------------
// Generic row-major GEMM, N fixed to 768: C[M x 768] = A[M x K] @ B[K x 768]
// Same 32-row / K-chunked structure; K must be a multiple of 192.
// ---------------------------------------------------------------------------
#define KCG 192
__global__ __launch_bounds__(256)
void gemm768(const float* __restrict__ A, const float* __restrict__ B,
             const float* __restrict__ bias, float* __restrict__ C, int K) {
    __shared__ float As[32][KCG + 4];      // stride 196: conflict-free

    const int m0 = blockIdx.x * 32;

    const int lane  = threadIdx.x & 31;
    const int wave  = threadIdx.x >> 5;
    const int mrow  = lane & 15;
    const int khalf = (lane >> 4) << 1;
    const int ncol  = lane & 15;
    const int mhalf = (lane >> 4) << 3;

    v8f acc0[6] = {};
    v8f acc1[6] = {};

    const int nch = K / KCG;
    for (int ch = 0; ch < nch; ch++) {
        __syncthreads();
        for (int i = threadIdx.x; i < 32 * KCG; i += 256) {
            int row = i / KCG;
            int col = i - row * KCG;
            As[row][col] = A[(size_t)(m0 + row) * K + ch * KCG + col];
        }
        __syncthreads();

        const int kbase = ch * KCG;
        for (int k0 = 0; k0 < KCG; k0 += 4) {
            v2f a0 = *(const v2f*)&As[mrow][k0 + khalf];
            v2f a1 = *(const v2f*)&As[mrow + 16][k0 + khalf];
            const float* brow = B + (size_t)(kbase + k0 + khalf) * DIMN + wave * 16 + ncol;
#pragma unroll
            for (int j = 0; j < 6; j++) {
                v2f b;
                b.x = brow[j * 128];
                b.y = brow[j * 128 + DIMN];
                acc0[j] = __builtin_amdgcn_wmma_f32_16x16x4_f32(
                    false, a0, false, b, (short)0, acc0[j], false, false);
                acc1[j] = __builtin_amdgcn_wmma_f32_16x16x4_f32(
                    false, a1, false, b, (short)0, acc1[j], false, false);
            }
        }
    }

#pragma unroll
    for (int j = 0; j < 6; j++) {
        int n    = j * 128 + wave * 16 + ncol;
        float bv = bias ? bias[n] : 0.0f;
#pragma unroll
        for (int r = 0; r < 8; r++) {
            int m = m0 + r + mhalf;
            C[(size_t)m * DIMN + n]        = acc0[j][r] + bv;
            C[(size_t)(m + 16) * DIMN + n] = acc1[j][r] + bv;
        }
    }
}

// ---------------------------------------------------------------------------
// Adjacent-pair routing dot: dots[b][i+1] = tokens[b][i] . U[b][i+1]
// sign(dots) == sign(reference cosine)  (norms are strictly positive)
// ---------------------------------------------------------------------------
__global__ __launch_bounds__(256)
void adj_dot(const float* __restrict__ tokens, const float* __restrict__ U,
             float* __restrict__ dots) {
    int b = blockIdx.y;
    int i = blockIdx.x;                    // 0..254
    const float* t = tokens + ((size_t)b * LTOK + i) * DIMN;
    const float* u = U      + ((size_t)b * LTOK + i + 1) * DIMN;

    float s = 0.0f;
    for (int e = threadIdx.x; e < DIMN; e += 256) s += t[e] * u[e];

    __shared__ float red[256];
    red[threadIdx.x] = s;
    __syncthreads();
    for (int off = 128; off > 0; off >>= 1) {
        if (threadIdx.x < off) red[threadIdx.x] += red[threadIdx.x + off];
        __syncthreads();
    }
    if (threadIdx.x == 0) dots[b * LTOK + i + 1] = red[0];
}

// ---------------------------------------------------------------------------
// Per-image compaction: scan boundary flags, stable gather, zero-pad, CLS row.
// boundary[l] = (l==0) || dots[b][l] < 0   (prob > 0.5  <=>  cos < 0)
// ---------------------------------------------------------------------------
__global__ __launch_bounds__(256)
void compact(const float* __restrict__ tokens, const float* __restrict__ dots,
             const float* __restrict__ cls, float* __restrict__ out, int max_len) {
    int b   = blockIdx.x;
    int tid = threadIdx.x;

    __shared__ int scan[256];
    __shared__ int order[256];

    int flag = (tid == 0) ? 1 : ((dots[b * LTOK + tid] < 0.0f) ? 1 : 0);
    scan[tid] = flag;
    __syncthreads();
    for (int off = 1; off < 256; off <<= 1) {
        int v = (tid >= off) ? scan[tid - off] : 0;
        __syncthreads();
        scan[tid] += v;
        __syncthreads();
    }
    int pos = scan[tid] - flag;            // exclusive prefix

    order[tid] = -1;
    __syncthreads();
    if (flag && pos < 256) order[pos] = tid;
    __syncthreads();

    const size_t rows = (size_t)max_len + 1;
    float* ob = out + (size_t)b * rows * DIMN;

    for (int e = tid; e < DIMN; e += 256) ob[e] = cls[e];   // CLS row

    for (int r = 1; r <= max_len; r++) {
        int src = order[r - 1];
        for (int e = tid; e < DIMN; e += 256) {
            float v = 0.0f;
            if (src >= 0) v = tokens[((size_t)b * LTOK + src) * DIMN + e];
            ob[(size_t)r * DIMN + e] = v;
        }
    }
}

// ---------------------------------------------------------------------------
extern "C" void kernel_launch(void* const* d_in, const int* in_sizes, int n_in,
                              void* d_out, int out_size, void* d_ws, size_t ws_size,
                              hipStream_t stream) {
    const float* x      = (const float*)d_in[0];   // [64,3,224,224]
    const float* conv_w = (const float*)d_in[1];   // [768,588]
    const float* conv_b = (const float*)d_in[2];   // [768]
    const float* cls    = (const float*)d_in[3];   // [768]
    const float* q_w    = (const float*)d_in[4];   // [768,768]
    const float* k_w    = (const float*)d_in[5];   // [768,768]
    float* out = (float*)d_out;

    // workspace layout (floats)
    float* Wt     = (float*)d_ws;                  // 588*768
    float* kt     = Wt     + KPATCH * DIMN;        // 768*768 (k_w transposed)
    float* G      = kt     + DIMN * DIMN;          // 768*768 (= k_w^T @ q_w)
    float* tokens = G      + DIMN * DIMN;          // 16384*768
    float* U      = tokens + (size_t)MTOT * DIMN;  // 16384*768 (= tokens @ G)
    float* dots   = U      + (size_t)MTOT * DIMN;  // 64*256

    const int max_len = out_size / (BATCH * DIMN) - 1;

    // 1) transposes
    transpose_k<<<(DIMN * KPATCH + 255) / 256, 256, 0, stream>>>(conv_w, Wt, DIMN, KPATCH);
    transpose_k<<<(DIMN * DIMN + 255) / 256, 256, 0, stream>>>(k_w, kt, DIMN, DIMN);

    // 2) patch embed (FP32 WMMA, 32-row blocks)
    patch_gemm<<<MTOT / 32, 256, 0, stream>>>(x, Wt, conv_b, tokens);

    // 3) routing bilinear form: G = k_w^T @ q_w ; U = tokens @ G  (FP32 WMMA)
    gemm768<<<DIMN / 32, 256, 0, stream>>>(kt, q_w, nullptr, G, DIMN);
    gemm768<<<MTOT / 32, 256, 0, stream>>>(tokens, G, nullptr, U, DIMN);

    // 4) adjacent cosine signs
    adj_dot<<<dim3(LTOK - 1, BATCH), 256, 0, stream>>>(tokens, U, dots);

    // 5) scan + stable compaction + CLS + zero padding
    compact<<<BATCH, 256, 0, stream>>>(tokens, dots, cls, out, max_len);
}